// TransformerBlock_48790828482875
// MI455X (gfx1250) — compile-verified
//
#include <hip/hip_runtime.h>
#include <math.h>

// Problem constants (from reference)
#define B_   2
#define N_   512
#define D_   128
#define E_   64
#define H_   8
#define HS_  16
#define FF_  512
#define EPS_ 1e-6f
#define LSTRIDE (N_ + 4)   // padded logits row to dodge LDS bank conflicts

typedef __attribute__((ext_vector_type(2))) float v2f;
typedef __attribute__((ext_vector_type(8))) float v8f;

// ---------------------------------------------------------------------------
// Kernel 1: per-(b,n) projections  q (pre-scaled), k, v, qe   (tiny, VALU)
// ---------------------------------------------------------------------------
__global__ __launch_bounds__(128) void prep_kernel(
    const float* __restrict__ node, const float* __restrict__ Wq,
    const float* __restrict__ Wk,   const float* __restrict__ Wv,
    float* __restrict__ q, float* __restrict__ k, float* __restrict__ v,
    float* __restrict__ qe)
{
  const int bn  = blockIdx.x;          // b*N + n
  const int tid = threadIdx.x;         // 0..127
  __shared__ float s_node[D_];
  __shared__ float s_q[H_ * HS_];

  s_node[tid] = node[(size_t)bn * D_ + tid];
  __syncthreads();

  const int h = tid >> 4, o = tid & 15;
  const float* wq = Wq + (size_t)(h * D_) * HS_ + o;          // Wq[h][i][o]
  const float* wk = Wk + (size_t)(h * (D_ + E_)) * HS_ + o;   // Wk[h][i][o]
  const float* wv = Wv + (size_t)(h * D_) * HS_ + o;
  float aq = 0.f, ak = 0.f, av = 0.f;
  #pragma unroll 4
  for (int i = 0; i < D_; ++i) {
    const float x = s_node[i];
    aq = fmaf(x, wq[i * HS_], aq);
    ak = fmaf(x, wk[i * HS_], ak);
    av = fmaf(x, wv[i * HS_], av);
  }
  aq *= 0.25f;                         // 1/sqrt(HS), HS=16
  const size_t idx = (size_t)bn * (H_ * HS_) + tid;
  q[idx] = aq; k[idx] = ak; v[idx] = av;
  s_q[tid] = aq;
  __syncthreads();

  // qe[h][e] = sum_o q[h][o] * Wk_edge[h][e][o]
  for (int he = tid; he < H_ * E_; he += 128) {
    const int hh = he >> 6, e = he & 63;
    const float* wke = Wk + (size_t)(hh * (D_ + E_) + D_ + e) * HS_;
    const float* qh  = s_q + hh * HS_;
    float a = 0.f;
    #pragma unroll
    for (int oo = 0; oo < HS_; ++oo) a = fmaf(qh[oo], wke[oo], a);
    qe[(size_t)bn * (H_ * E_) + he] = a;
  }
}

// ---------------------------------------------------------------------------
// Kernel 2: fused attention + projection + LN1 + FFN + LN2, one WG per (b,n)
// ---------------------------------------------------------------------------
__global__ __launch_bounds__(256) void attn_block_kernel(
    const float* __restrict__ node, const float* __restrict__ edge,
    const float* __restrict__ q,    const float* __restrict__ k,
    const float* __restrict__ v,    const float* __restrict__ qe,
    const float* __restrict__ Wp,   const float* __restrict__ bp,
    const float* __restrict__ ln1_g,const float* __restrict__ ln1_b,
    const float* __restrict__ Wf1,  const float* __restrict__ bf1,
    const float* __restrict__ Wf2,  const float* __restrict__ bf2,
    const float* __restrict__ ln2_g,const float* __restrict__ ln2_b,
    float* __restrict__ out)
{
  const int bn   = blockIdx.x;             // b*N + n
  const int tid  = threadIdx.x;
  const int lane = tid & 31;
  const int wave = tid >> 5;
  const int b    = bn >> 9;                // N_ == 512
  const size_t bRow = (size_t)b * N_;

  __shared__ float s_q[D_];
  __shared__ float s_logits[H_ * LSTRIDE];
  __shared__ float s_red[16];
  __shared__ float s_mh[D_];
  __shared__ float s_x[D_];
  __shared__ float s_ff[FF_];

  if (tid < D_) s_q[tid] = q[(size_t)bn * D_ + tid];

  // --- Build B-fragments (qe^T, 64e x 16h padded) for V_WMMA_F32_16X16X4_F32.
  // ISA layout (mirrors A): v0 = row K(+0) lanes 0-15 / row K(+2) lanes 16-31,
  // v1 = row K(+1) / K(+3); column N = lane&15 (head, zero-padded for h>=8).
  const int col  = lane & 15;
  const int koff = (lane >> 4) << 1;       // 0 or 2
  const float* qe_bn = qe + (size_t)bn * (H_ * E_);
  v2f bfrag[16];
  #pragma unroll
  for (int s = 0; s < 16; ++s) {
    const int e0 = 4 * s + koff;
    float bx = 0.f, by = 0.f;
    if (col < H_) { bx = qe_bn[col * E_ + e0]; by = qe_bn[col * E_ + e0 + 1]; }
    bfrag[s].x = bx; bfrag[s].y = by;
  }
  __syncthreads();                          // s_q visible

  // q row for this lane's head (clamped for padded lanes; results unused)
  const int hc = col & (H_ - 1);
  float qreg[16];
  #pragma unroll
  for (int oo = 0; oo < 16; ++oo) qreg[oo] = (col < H_) ? s_q[hc * HS_ + oo] : 0.f;

  // --- Stream edge slice: logits[m, h] = edge[m,:] . qe[h,:]  (+ q.k dot) ---
  const float* e_bn = edge + (size_t)bn * N_ * E_;
  const int arow = lane & 15;               // A-matrix M row for this lane
  for (int t = wave; t < N_ / 16; t += 8) {
    const int m0 = t * 16;
    const float* rowp = e_bn + (size_t)(m0 + arow) * E_ + koff;
    if (t + 8 < N_ / 16) __builtin_prefetch(rowp + 128 * E_, 0, 1);
    v8f acc = {};
    #pragma unroll
    for (int s = 0; s < 16; ++s) {
      const v2f a = *(const v2f*)(rowp + 4 * s);
      acc = __builtin_amdgcn_wmma_f32_16x16x4_f32(
          /*neg_a=*/false, a, /*neg_b=*/false, bfrag[s],
          /*c_mod=*/(short)0, acc, /*reuse_a=*/false, /*reuse_b=*/false);
    }
    // C layout: lane (g=lane>>4, col) ; VGPR r -> M = r + 8*g, N = col
    const int g = lane >> 4;
    #pragma unroll
    for (int r = 0; r < 8; ++r) {
      const int m = m0 + r + 8 * g;
      const float4* kp4 =
          (const float4*)(k + ((bRow + m) * H_ + hc) * HS_);
      float dot = 0.f;
      #pragma unroll
      for (int oq = 0; oq < 4; ++oq) {
        const float4 kk = kp4[oq];
        dot = fmaf(qreg[4 * oq + 0], kk.x, dot);
        dot = fmaf(qreg[4 * oq + 1], kk.y, dot);
        dot = fmaf(qreg[4 * oq + 2], kk.z, dot);
        dot = fmaf(qreg[4 * oq + 3], kk.w, dot);
      }
      if (col < H_) s_logits[col * LSTRIDE + m] = acc[r] + dot;
    }
  }
  __syncthreads();

  // --- Per-head softmax + attn@v : wave == head -------------------------------
  {
    const int hh = wave;
    float vals[16];
    float mx = -3.402823466e38f;
    #pragma unroll
    for (int i = 0; i < 16; ++i) {
      vals[i] = s_logits[hh * LSTRIDE + lane + 32 * i];
      mx = fmaxf(mx, vals[i]);
    }
    #pragma unroll
    for (int off = 16; off >= 1; off >>= 1) mx = fmaxf(mx, __shfl_xor(mx, off, 32));
    float sum = 0.f;
    #pragma unroll
    for (int i = 0; i < 16; ++i) { vals[i] = __expf(vals[i] - mx); sum += vals[i]; }
    #pragma unroll
    for (int off = 16; off >= 1; off >>= 1) sum += __shfl_xor(sum, off, 32);
    const float inv = 1.f / sum;

    float acc_o[16];
    #pragma unroll
    for (int o = 0; o < 16; ++o) acc_o[o] = 0.f;
    for (int i = 0; i < 16; ++i) {
      const int m = lane + 32 * i;
      const float a = vals[i] * inv;
      const float4* vp4 = (const float4*)(v + ((bRow + m) * H_ + hh) * HS_);
      #pragma unroll
      for (int oq = 0; oq < 4; ++oq) {
        const float4 vv = vp4[oq];
        acc_o[4 * oq + 0] = fmaf(a, vv.x, acc_o[4 * oq + 0]);
        acc_o[4 * oq + 1] = fmaf(a, vv.y, acc_o[4 * oq + 1]);
        acc_o[4 * oq + 2] = fmaf(a, vv.z, acc_o[4 * oq + 2]);
        acc_o[4 * oq + 3] = fmaf(a, vv.w, acc_o[4 * oq + 3]);
      }
    }
    #pragma unroll
    for (int off = 16; off >= 1; off >>= 1) {
      #pragma unroll
      for (int o = 0; o < 16; ++o) acc_o[o] += __shfl_xor(acc_o[o], off, 32);
    }
    if (lane == 0) {
      #pragma unroll
      for (int o = 0; o < 16; ++o) s_mh[hh * HS_ + o] = acc_o[o];
    }
  }
  __syncthreads();

  // --- Output projection + residual + LN1 ------------------------------------
  float tval = 0.f;
  if (tid < D_) {
    float a = bp[tid];
    #pragma unroll 4
    for (int ho = 0; ho < D_; ++ho) a = fmaf(s_mh[ho], Wp[(size_t)ho * D_ + tid], a);
    tval = node[(size_t)bn * D_ + tid] + a;
  }
  float p = tval, p2 = tval * tval;
  #pragma unroll
  for (int off = 16; off >= 1; off >>= 1) {
    p += __shfl_xor(p, off, 32); p2 += __shfl_xor(p2, off, 32);
  }
  if (lane == 0) { s_red[wave] = p; s_red[8 + wave] = p2; }
  __syncthreads();
  float mu = 0.f, ms = 0.f;
  #pragma unroll
  for (int w = 0; w < 8; ++w) { mu += s_red[w]; ms += s_red[8 + w]; }
  mu *= (1.f / D_); ms *= (1.f / D_);
  float rstd = rsqrtf(ms - mu * mu + EPS_);
  __syncthreads();
  if (tid < D_) s_x[tid] = (tval - mu) * rstd * ln1_g[tid] + ln1_b[tid];
  __syncthreads();

  // --- FFN: relu(x @ Wf1 + bf1) @ Wf2 + bf2 ----------------------------------
  for (int j = tid; j < FF_; j += 256) {
    float a = bf1[j];
    #pragma unroll 4
    for (int i = 0; i < D_; ++i) a = fmaf(s_x[i], Wf1[(size_t)i * FF_ + j], a);
    s_ff[j] = fmaxf(a, 0.f);
  }
  __syncthreads();

  float yval = 0.f;
  if (tid < D_) {
    float a = bf2[tid];
    #pragma unroll 4
    for (int j = 0; j < FF_; ++j) a = fmaf(s_ff[j], Wf2[(size_t)j * D_ + tid], a);
    yval = s_x[tid] + a;
  }
  p = yval; p2 = yval * yval;
  #pragma unroll
  for (int off = 16; off >= 1; off >>= 1) {
    p += __shfl_xor(p, off, 32); p2 += __shfl_xor(p2, off, 32);
  }
  if (lane == 0) { s_red[wave] = p; s_red[8 + wave] = p2; }
  __syncthreads();
  mu = 0.f; ms = 0.f;
  #pragma unroll
  for (int w = 0; w < 8; ++w) { mu += s_red[w]; ms += s_red[8 + w]; }
  mu *= (1.f / D_); ms *= (1.f / D_);
  rstd = rsqrtf(ms - mu * mu + EPS_);
  if (tid < D_) out[(size_t)bn * D_ + tid] = (yval - mu) * rstd * ln2_g[tid] + ln2_b[tid];
}

// ---------------------------------------------------------------------------
extern "C" void kernel_launch(void* const* d_in, const int* in_sizes, int n_in,
                              void* d_out, int out_size, void* d_ws, size_t ws_size,
                              hipStream_t stream) {
  (void)in_sizes; (void)n_in; (void)out_size; (void)ws_size;
  const float* node  = (const float*)d_in[0];
  const float* edge  = (const float*)d_in[1];
  const float* Wq    = (const float*)d_in[2];
  const float* Wk    = (const float*)d_in[3];
  const float* Wv    = (const float*)d_in[4];
  const float* Wp    = (const float*)d_in[5];
  const float* bp    = (const float*)d_in[6];
  const float* ln1_g = (const float*)d_in[7];
  const float* ln1_b = (const float*)d_in[8];
  const float* Wf1   = (const float*)d_in[9];
  const float* bf1   = (const float*)d_in[10];
  const float* Wf2   = (const float*)d_in[11];
  const float* bf2   = (const float*)d_in[12];
  const float* ln2_g = (const float*)d_in[13];
  const float* ln2_b = (const float*)d_in[14];
  float* out = (float*)d_out;

  float* ws = (float*)d_ws;
  float* q  = ws;                 // B*N*H*HS = 131072 floats
  float* k  = q  + (size_t)B_ * N_ * H_ * HS_;
  float* v  = k  + (size_t)B_ * N_ * H_ * HS_;
  float* qe = v  + (size_t)B_ * N_ * H_ * HS_;   // B*N*H*E = 524288 floats

  prep_kernel<<<B_ * N_, 128, 0, stream>>>(node, Wq, Wk, Wv, q, k, v, qe);
  attn_block_kernel<<<B_ * N_, 256, 0, stream>>>(node, edge, q, k, v, qe,
      Wp, bp, ln1_g, ln1_b, Wf1, bf1, Wf2, bf2, ln2_g, ln2_b, out);
}